// SpatialTransformerGauss2d_9792525435345
// MI455X (gfx1250) — compile-verified
//
#include <hip/hip_runtime.h>
#include <cstdint>

typedef unsigned int u32;
typedef unsigned long long u64;
typedef u32 v4u __attribute__((ext_vector_type(4)));
typedef int  v8i __attribute__((ext_vector_type(8)));
typedef int  v4i __attribute__((ext_vector_type(4)));

#define C2      1024      // N*ch = 64*16 columns per pixel-row
#define P_TOT   8192
#define KTOT    80
#define ROWS_TOTAL 21824  // 128^2 + 64^2 + 32^2 + 16^2 + 8^2
#define PPB     4         // points per workgroup (double buffered)

// ---------------------------------------------------------------------------
// Stage 1: fused depthwise 3x3 Gaussian (zero pad) + Laplacian + decimate.
// Reads NCHW (coalesced along x), writes Laplacian in pixel-major [pix][nc]
// layout via an LDS transpose tile, writes decimated next level in NCHW.
// grid = (W/8, H, 32), block = 256: thread = (nc_i = t>>3 in 0..31, x_i = t&7)
// ---------------------------------------------------------------------------
__global__ __launch_bounds__(256)
void pyr_kernel(const float* __restrict__ src,
                float* __restrict__ rows, int rowBase,
                float* __restrict__ nxt,
                int H, int W) {
    __shared__ float tile[8 * 33];
    const int t   = threadIdx.x;
    const int xi  = t & 7;
    const int nci = t >> 3;
    const int nc  = blockIdx.z * 32 + nci;
    const int x   = blockIdx.x * 8 + xi;
    const int y   = blockIdx.y;

    const float* plane = src + (size_t)nc * H * W;
    float s = 0.f;
    const float c = plane[y * W + x];
#pragma unroll
    for (int dy = -1; dy <= 1; ++dy) {
        const int yy = y + dy;
#pragma unroll
        for (int dx = -1; dx <= 1; ++dx) {
            const int xx = x + dx;
            const float k = ((dy == 0) ? 2.f : 1.f) * ((dx == 0) ? 2.f : 1.f) * 0.0625f;
            if (yy >= 0 && yy < H && xx >= 0 && xx < W)
                s += k * plane[yy * W + xx];
        }
    }
    const float lap = c - s;
    tile[xi * 33 + nci] = lap;

    // decimated low-pass -> next level (even rows/cols), NCHW
    if (((y & 1) == 0) && ((x & 1) == 0))
        nxt[(size_t)nc * (H >> 1) * (W >> 1) + (y >> 1) * (W >> 1) + (x >> 1)] = s;

    __syncthreads();
    const int nc2 = t & 31;
    const int x2  = t >> 5;
    rows[(size_t)(rowBase + y * W + blockIdx.x * 8 + x2) * C2 +
         blockIdx.z * 32 + nc2] = tile[x2 * 33 + nc2];
}

// Final 8x8 base image NCHW -> pixel-major rows (tiny: 256KB total)
__global__ __launch_bounds__(256)
void lastlevel_kernel(const float* __restrict__ img4,
                      float* __restrict__ rows, int rowBase) {
    const int pix = blockIdx.x;
    const int t = threadIdx.x;
#pragma unroll
    for (int k = 0; k < 4; ++k) {
        const int nc = t + k * 256;
        rows[(size_t)(rowBase + pix) * C2 + nc] = img4[(size_t)nc * 64 + pix];
    }
}

// features (1,80,1,P) -> featT[p*80 + k] for contiguous per-point reads
__global__ __launch_bounds__(256)
void feat_tr_kernel(const float* __restrict__ f, float* __restrict__ ft) {
    const int g = blockIdx.x * 256 + threadIdx.x;
    if (g < KTOT * P_TOT) {
        const int k = g % KTOT;
        const int p = g / KTOT;
        ft[g] = f[(size_t)k * P_TOT + p];
    }
}

// ---------------------------------------------------------------------------
// Stage 2: TDM-gather bilinear sampling + channel contraction.
// grid = (P/PPB, 4): blockIdx.y = batch-quarter (16 n values -> 256 columns).
// Per point: 20 rows (5 levels x 4 corners) x 1KB gathered into LDS by the
// Tensor Data Mover (gather mode, 16-bit row indices), double buffered.
// Thread t = (n_local = t>>4, ch = t&15); 16-lane shuffle-reduce over ch.
// ---------------------------------------------------------------------------
__global__ __launch_bounds__(256)
void sample_kernel(const float* __restrict__ rows,
                   const float* __restrict__ grid_pts,
                   const float* __restrict__ featT,
                   const float* __restrict__ bias,
                   float* __restrict__ out) {
    __shared__ float buf[2][20][256];
    const int t  = threadIdx.x;
    const int q  = blockIdx.y;          // batch quarter
    const int ch = t & 15;
    const int nl = t >> 4;
    const int pbase = blockIdx.x * PPB;

    // global byte address of this quarter's column window within each row
    const u64 gbase = (u64)(uintptr_t)rows + (u64)q * (256u * 4u);

    auto meta = [&](int p, int* ri, float* wv) {
        float gx = grid_pts[2 * p + 0];
        float gy = grid_pts[2 * p + 1];
        gx = fminf(fmaxf(gx, -1.f), 1.f);
        gy = fminf(fmaxf(gy, -1.f), 1.f);
        const int dims[5] = {128, 64, 32, 16, 8};
        const int base[5] = {0, 16384, 20480, 21504, 21760};
#pragma unroll
        for (int l = 0; l < 5; ++l) {
            const int D = dims[l];
            const float ix = ((gx + 1.f) * D - 1.f) * 0.5f;
            const float iy = ((gy + 1.f) * D - 1.f) * 0.5f;
            const float ix0f = floorf(ix), iy0f = floorf(iy);
            const float fx = ix - ix0f, fy = iy - iy0f;
            const int ix0 = (int)ix0f, iy0 = (int)iy0f;
            const int ix1 = ix0 + 1,  iy1 = iy0 + 1;
            const float vx0 = (ix0 >= 0 && ix0 < D) ? 1.f : 0.f;
            const float vx1 = (ix1 >= 0 && ix1 < D) ? 1.f : 0.f;
            const float vy0 = (iy0 >= 0 && iy0 < D) ? 1.f : 0.f;
            const float vy1 = (iy1 >= 0 && iy1 < D) ? 1.f : 0.f;
            const int cx0 = min(max(ix0, 0), D - 1), cx1 = min(max(ix1, 0), D - 1);
            const int cy0 = min(max(iy0, 0), D - 1), cy1 = min(max(iy1, 0), D - 1);
            ri[4 * l + 0] = base[l] + cy0 * D + cx0;
            ri[4 * l + 1] = base[l] + cy0 * D + cx1;
            ri[4 * l + 2] = base[l] + cy1 * D + cx0;
            ri[4 * l + 3] = base[l] + cy1 * D + cx1;
            wv[4 * l + 0] = (1.f - fy) * (1.f - fx) * vy0 * vx0;
            wv[4 * l + 1] = (1.f - fy) * fx * vy0 * vx1;
            wv[4 * l + 2] = fy * (1.f - fx) * vy1 * vx0;
            wv[4 * l + 3] = fy * fx * vy1 * vx1;
        }
    };

    auto pk = [](int a, int b) -> int {
        return (int)(((u32)a & 0xFFFFu) | (((u32)b & 0xFFFFu) << 16));
    };

    // Issue two TDM gathers into buffer b: rows 0..15 (levels 0-3), 16..19 (level 4).
    // D# per CDNA5 ISA: gather_mode=1, 16-bit indices,
    // data_size=4B, tensor_dim0=tile_dim0=256 elems, dim0_stride=1024 elems,
    // tensor_dim1=21824 rows, type=2. clang-23 6-arg form: extra zero group.
    auto issue = [&](int b, const int* ri) {
        const u32 gaLo = (u32)gbase;
        const u32 gaHi = ((u32)(gbase >> 32) & 0x01FFFFFFu) | 0x80000000u; // type=2
        const v8i gz = {0, 0, 0, 0, 0, 0, 0, 0};
        {
            const u32 lds0 = (u32)(uintptr_t)&buf[b][0][0];
            v4u g0 = {0x80000001u, lds0, gaLo, gaHi};
            v8i g1 = {0x00020000, 0x01000000, 0x55400000, 0x01000000, 16, 1024, 0, 0};
            v4i g2 = {pk(ri[0], ri[1]), pk(ri[2], ri[3]), pk(ri[4], ri[5]), pk(ri[6], ri[7])};
            v4i g3 = {pk(ri[8], ri[9]), pk(ri[10], ri[11]), pk(ri[12], ri[13]), pk(ri[14], ri[15])};
            __builtin_amdgcn_tensor_load_to_lds(g0, g1, g2, g3, gz, 0);
        }
        {
            const u32 lds1 = (u32)(uintptr_t)&buf[b][16][0];
            v4u g0 = {0x80000001u, lds1, gaLo, gaHi};
            v8i g1 = {0x00020000, 0x01000000, 0x55400000, 0x01000000, 4, 1024, 0, 0};
            v4i g2 = {pk(ri[16], ri[17]), pk(ri[18], ri[19]), 0, 0};
            v4i g3 = {0, 0, 0, 0};
            __builtin_amdgcn_tensor_load_to_lds(g0, g1, g2, g3, gz, 0);
        }
    };

    int   rowsN[20];
    float wN[20];

    meta(pbase, rowsN, wN);
    if (t < 32) {                       // wave 0 only: TDM issues ignore EXEC
        issue(0, rowsN);
        __builtin_amdgcn_s_wait_tensorcnt(0);
    }
    __syncthreads();

#pragma unroll 1
    for (int i = 0; i < PPB; ++i) {
        const int p = pbase + i;
        float wC[20];
#pragma unroll
        for (int j = 0; j < 20; ++j) wC[j] = wN[j];

        if (i + 1 < PPB) {
            meta(p + 1, rowsN, wN);     // all threads: weights needed next iter
            if (t < 32) issue((i + 1) & 1, rowsN);
        }

        float fv[5];
#pragma unroll
        for (int l = 0; l < 5; ++l)
            fv[l] = featT[(size_t)p * KTOT + l * 16 + ch];

        float acc = 0.f;
#pragma unroll
        for (int j = 0; j < 20; ++j)
            acc += buf[i & 1][j][t] * (wC[j] * fv[j >> 2]);

        // reduce over 16 channel lanes
        acc += __shfl_xor(acc, 1, 32);
        acc += __shfl_xor(acc, 2, 32);
        acc += __shfl_xor(acc, 4, 32);
        acc += __shfl_xor(acc, 8, 32);
        if (ch == 0)
            out[(size_t)(q * 16 + nl) * P_TOT + p] = acc + bias[p];

        if (t < 32) __builtin_amdgcn_s_wait_tensorcnt(0);
        __syncthreads();
    }
}

// ---------------------------------------------------------------------------
extern "C" void kernel_launch(void* const* d_in, const int* in_sizes, int n_in,
                              void* d_out, int out_size, void* d_ws, size_t ws_size,
                              hipStream_t stream) {
    (void)in_sizes; (void)n_in; (void)out_size; (void)ws_size;
    const float* x        = (const float*)d_in[0];
    const float* grid_pts = (const float*)d_in[1];
    const float* features = (const float*)d_in[2];
    const float* bias     = (const float*)d_in[3];
    float* out = (float*)d_out;

    float* ws    = (float*)d_ws;
    float* rows  = ws;                                  // 21824*1024 = 22,347,776
    float* img1  = rows + (size_t)ROWS_TOTAL * C2;      // 1024*64*64 = 4,194,304
    float* img2  = img1 + (size_t)1024 * 64 * 64;       // 1024*32*32 = 1,048,576
    float* img3  = img2 + (size_t)1024 * 32 * 32;       // 1024*16*16 =   262,144
    float* img4  = img3 + (size_t)1024 * 16 * 16;       // 1024*8*8   =    65,536
    float* featT = img4 + (size_t)1024 * 8 * 8;         // 80*8192    =   655,360

    pyr_kernel<<<dim3(16, 128, 32), 256, 0, stream>>>(x,    rows, 0,     img1, 128, 128);
    pyr_kernel<<<dim3( 8,  64, 32), 256, 0, stream>>>(img1, rows, 16384, img2,  64,  64);
    pyr_kernel<<<dim3( 4,  32, 32), 256, 0, stream>>>(img2, rows, 20480, img3,  32,  32);
    pyr_kernel<<<dim3( 2,  16, 32), 256, 0, stream>>>(img3, rows, 21504, img4,  16,  16);
    lastlevel_kernel<<<64, 256, 0, stream>>>(img4, rows, 21760);
    feat_tr_kernel<<<(KTOT * P_TOT + 255) / 256, 256, 0, stream>>>(features, featT);
    sample_kernel<<<dim3(P_TOT / PPB, 4), 256, 0, stream>>>(rows, grid_pts, featT, bias, out);
}